// SineModel_15556371546374
// MI455X (gfx1250) — compile-verified
//
#include <hip/hip_runtime.h>

// Problem constants (from reference)
#define NF    16
#define NC    1
#define T     256
#define BATCH 32
#define BETA  2.5f
#define W_BI  0.05f
#define W_L2  0.01f
#define EPSW  1e-12f

#define AS_STRIDE 516   // 512 + 4 dword pad -> conflict-free b64 fragment loads

typedef __attribute__((ext_vector_type(2))) float v2f;
typedef __attribute__((ext_vector_type(8))) float v8f;

__device__ __forceinline__ float clip01(float v) { return fminf(fmaxf(v, 0.f), 1.f); }
__device__ __forceinline__ float sigm(float z)   { return 1.f / (1.f + __expf(-z)); }

// ---------------------------------------------------------------------------
// Kernel 1: prep
//  blocks [0, 512):  per (b,f) compute V = a*x - b, global vmax/vmin,
//                    store 4 signals: E*V, E, Em*V, Em   (E=exp(BETA(V-vmax)))
//  blocks [512,528): per f compute window tables w1/w2 (+EPSW folded in)
// ---------------------------------------------------------------------------
__global__ __launch_bounds__(256)
void prep_kernel(const float* __restrict__ x,
                 const float* __restrict__ t1,
                 const float* __restrict__ t2,
                 const float* __restrict__ av,
                 const float* __restrict__ bv,
                 float* __restrict__ sig,    // [B][NF][4][T]
                 float* __restrict__ wtab1,  // [NF][T]
                 float* __restrict__ wtab2)  // [NF][T]
{
    __shared__ float red[256];
    const int blk = blockIdx.x;
    const int t   = threadIdx.x;

    if (blk < BATCH * NF) {
        const int b = blk >> 4, f = blk & 15;
        const float V = av[f] * x[(b * T + t) * 2] - bv[f];

        red[t] = V; __syncthreads();
        for (int s = 128; s > 0; s >>= 1) { if (t < s) red[t] = fmaxf(red[t], red[t + s]); __syncthreads(); }
        const float vmax = red[0]; __syncthreads();
        red[t] = V; __syncthreads();
        for (int s = 128; s > 0; s >>= 1) { if (t < s) red[t] = fminf(red[t], red[t + s]); __syncthreads(); }
        const float vmin = red[0];

        const float E  = __expf( BETA * (V - vmax));   // <= 1
        const float Em = __expf(-BETA * (V - vmin));   // <= 1
        float* sp = sig + (size_t)((b * NF + f) * 4) * T;
        sp[0 * T + t] = E  * V;
        sp[1 * T + t] = E;
        sp[2 * T + t] = Em * V;
        sp[3 * T + t] = Em;
    } else {
        const int f = blk - BATCH * NF;
        const float j = (float)t;
        // straight-through forward clipping of t1/t2 (both rows)
        float t1h0 = fminf(fmaxf(t1[f], 0.f), (float)(T - 1));
        float t2h0 = fminf(fmaxf(t2[f], 0.f), (float)(T - 1));
        if (t1h0 >= t2h0 - 1.f) t1h0 = t2h0 - 1.f;
        wtab1[f * T + t] = sigm(BETA * (j - t1h0)) * sigm(BETA * (t2h0 - j)) + EPSW;

        float t1h1 = fminf(fmaxf(t1[NF + f], 0.f), (float)(T - 1));
        float t2h1 = fminf(fmaxf(t2[NF + f], 0.f), (float)(T - 1));
        if (t1h1 >= t2h1 - 1.f) t1h1 = t2h1 - 1.f;
        wtab2[f * T + t] = sigm(BETA * (j - t1h1)) * sigm(BETA * (t2h1 - j)) + EPSW;
    }
}

// ---------------------------------------------------------------------------
// Kernel 2: stage-1 as banded-Toeplitz GEMM on V_WMMA_F32_16X16X4_F32.
// Per feature f: Y(128x256) = G(128x511) x H(511x256),
//   rows = (b,r) with r in {E*V, E, Em*V, Em},  H[k,i] = w1_f[k-i] (band 0..255),
//   padded signal G[k] = sig[min(k,255)].
//
// Block = (f, m-tile): stages its 16 signal rows (pre-clamped to length 512)
// and the zero-padded window table into LDS once; 16 waves each own one
// 16-wide n-tile.  Inner loop: unconditional ds_load_b64 (A) + 2x ds_load_b32
// (B, zero padding handles the band) + v_wmma_f32_16x16x4_f32.  No exec
// masking, no global traffic, no address clamps in the loop.
// ---------------------------------------------------------------------------
__global__ __launch_bounds__(512)
void stage1_wmma_kernel(const float* __restrict__ sig,    // [B][NF][4][T]
                        const float* __restrict__ wtab1,  // [NF][T]
                        const float* __restrict__ tvt,    // (2,NF)
                        float* __restrict__ r1i)          // [B][NF][T]
{
    __shared__ float As[16 * AS_STRIDE];   // 16 rows x 512 padded signal (+pad)
    __shared__ float wpad[304];            // wpad[16+d] = w[d] for d in [0,256), else 0
    const int blk = blockIdx.x;            // 128 blocks = 16 f x 8 m-tiles
    const int f   = blk >> 3;
    const int mt  = blk & 7;
    const int tid = threadIdx.x;           // 512 threads = 16 waves
    const int m0  = mt * 16;

    // ---- stage window table with zero halo (d in [-16, 288)) ----
    for (int idx = tid; idx < 304; idx += 512) {
        const int d = idx - 16;
        wpad[idx] = (d >= 0 && d < T) ? wtab1[f * T + d] : 0.f;
    }
    // ---- stage 16 signal rows, tail-clamped to length 512 ----
    for (int e = tid; e < 16 * 512; e += 512) {
        const int r = e >> 9;              // row 0..15 (one row per iteration)
        const int k = e & 511;             // 0..511
        const int grow = m0 + r;
        const int bb = grow >> 2, rr = grow & 3;
        As[r * AS_STRIDE + k] =
            sig[(size_t)((bb * NF + f) * 4 + rr) * T + (k <= T - 1 ? k : T - 1)];
    }
    __syncthreads();

    const int nt = tid >> 5;               // wave id = n-tile 0..15
    const int L  = tid & 31;
    const int i0 = nt * 16;

    // A fragment: lane L holds row m = L&15, K pair selected by L>>4
    const int m     = L & 15;
    const int khalf = (L >> 4) << 1;       // 0 or 2
    // B fragment: lane L holds col n = L&15, same K-half split
    const int n = L & 15;
    const int i = i0 + n;

    const float* arow = As + m * AS_STRIDE;

    v8f c = {};
    #pragma unroll 4
    for (int kb = i0; kb < i0 + 272; kb += 4) {
        const int ka = kb + khalf;
        v2f a;
        a.x = arow[ka];
        a.y = arow[ka + 1];
        const int idx = 16 + ka - i;       // in [1, 288): always in wpad
        v2f bf;
        bf.x = wpad[idx];
        bf.y = wpad[idx + 1];
        c = __builtin_amdgcn_wmma_f32_16x16x4_f32(
                /*neg_a=*/false, a, /*neg_b=*/false, bf,
                /*c_mod=*/(short)0, c, /*reuse_a=*/false, /*reuse_b=*/false);
    }

    // C layout: lane L holds rows v + 8*(L>>4) in c[v]; 4 rows per batch entry.
    const float tvt0 = clip01(tvt[f]);
    const int bb0 = (m0 >> 2) + ((L >> 4) << 1);
    const float F1a = c[0] / c[1], G1a = c[2] / c[3];
    const float F1b = c[4] / c[5], G1b = c[6] / c[7];
    r1i[(size_t)((bb0)     * NF + f) * T + i] = tvt0 * F1a + (1.f - tvt0) * G1a;
    r1i[(size_t)((bb0 + 1) * NF + f) * T + i] = tvt0 * F1b + (1.f - tvt0) * G1b;
}

// ---------------------------------------------------------------------------
// Kernel 3: stage 2 (soft max/min over T with w2) + stage 3 (over NF with p)
// One block per batch element; wave f reduces its 256-long row with shuffles.
// Block 0 also emits the bi+l2 penalty scalar.
// ---------------------------------------------------------------------------
__global__ __launch_bounds__(512)
void finalize_kernel(const float* __restrict__ r1i,    // [B][NF][T]
                     const float* __restrict__ wtab2,  // [NF][T]
                     const float* __restrict__ tvt,    // (2,NF)
                     const float* __restrict__ tvl,    // (1,)
                     const float* __restrict__ p,      // (NC,NF)
                     float* __restrict__ out)          // [BATCH+1]
{
    __shared__ float r1s[NF];
    const int b   = blockIdx.x;
    const int tid = threadIdx.x;
    const int f   = tid >> 5;
    const int L   = tid & 31;

    const float* rp = r1i + (size_t)(b * NF + f) * T;
    const float* wp = wtab2 + f * T;

    float v[8], wv[8];
    float vmax = -3.4e38f, vmin = 3.4e38f;
    #pragma unroll
    for (int j = 0; j < 8; j++) {
        const int idx = L + 32 * j;
        v[j] = rp[idx]; wv[j] = wp[idx];
        vmax = fmaxf(vmax, v[j]); vmin = fminf(vmin, v[j]);
    }
    for (int msk = 16; msk > 0; msk >>= 1) {
        vmax = fmaxf(vmax, __shfl_xor(vmax, msk, 32));
        vmin = fminf(vmin, __shfl_xor(vmin, msk, 32));
    }
    float nh = 0.f, dh = 0.f, nl = 0.f, dl = 0.f;
    #pragma unroll
    for (int j = 0; j < 8; j++) {
        const float eh = wv[j] * __expf( BETA * (v[j] - vmax));
        const float el = wv[j] * __expf(-BETA * (v[j] - vmin));
        nh += eh * v[j]; dh += eh;
        nl += el * v[j]; dl += el;
    }
    for (int msk = 16; msk > 0; msk >>= 1) {
        nh += __shfl_xor(nh, msk, 32); dh += __shfl_xor(dh, msk, 32);
        nl += __shfl_xor(nl, msk, 32); dl += __shfl_xor(dl, msk, 32);
    }
    if (L == 0) {
        const float tvt1 = clip01(tvt[NF + f]);
        r1s[f] = tvt1 * (nh / dh) + (1.f - tvt1) * (nl / dl);
    }
    __syncthreads();

    if (tid == 0) {
        const float tl = clip01(tvl[0]);
        float rmax = -3.4e38f, rmin = 3.4e38f;
        float pcv[NF];
        for (int ff = 0; ff < NF; ff++) {
            pcv[ff] = clip01(p[ff]) + EPSW;
            rmax = fmaxf(rmax, r1s[ff]); rmin = fminf(rmin, r1s[ff]);
        }
        float nh2 = 0.f, dh2 = 0.f, nl2 = 0.f, dl2 = 0.f;
        for (int ff = 0; ff < NF; ff++) {
            const float vv = r1s[ff];
            const float eh = pcv[ff] * __expf( BETA * (vv - rmax));
            const float el = pcv[ff] * __expf(-BETA * (vv - rmin));
            nh2 += eh * vv; dh2 += eh;
            nl2 += el * vv; dl2 += el;
        }
        out[b] = tl * (nh2 / dh2) + (1.f - tl) * (nl2 / dl2);

        if (b == 0) {
            float bi = 0.f, l2 = 0.f;
            for (int k = 0; k < 2 * NF; k++) { const float tc = clip01(tvt[k]); bi += tc * (1.f - tc); }
            { const float tc = clip01(tvl[0]); bi += tc * (1.f - tc); }
            for (int k = 0; k < NC * NF; k++) { const float pc = clip01(p[k]); bi += pc * (1.f - pc); l2 += pc * pc; }
            out[BATCH] = W_BI * bi + W_L2 * l2;
        }
    }
}

// ---------------------------------------------------------------------------
extern "C" void kernel_launch(void* const* d_in, const int* in_sizes, int n_in,
                              void* d_out, int out_size, void* d_ws, size_t ws_size,
                              hipStream_t stream)
{
    (void)in_sizes; (void)n_in; (void)out_size; (void)ws_size;
    const float* x   = (const float*)d_in[0];  // (32,256,2)
    const float* t1  = (const float*)d_in[1];  // (2,16)
    const float* t2  = (const float*)d_in[2];  // (2,16)
    const float* av  = (const float*)d_in[3];  // (16,)
    const float* bv  = (const float*)d_in[4];  // (16,)
    const float* p   = (const float*)d_in[5];  // (1,16)
    const float* tvt = (const float*)d_in[6];  // (2,16)
    const float* tvl = (const float*)d_in[7];  // (1,)
    float* out = (float*)d_out;                // 32 + 1

    float* ws    = (float*)d_ws;
    float* sig   = ws;                          // 32*16*4*256 = 524288 f32
    float* wtab1 = sig   + BATCH * NF * 4 * T;  // 4096 f32
    float* wtab2 = wtab1 + NF * T;              // 4096 f32
    float* r1i   = wtab2 + NF * T;              // 131072 f32  (~2.6 MB total)

    prep_kernel<<<BATCH * NF + NF, 256, 0, stream>>>(x, t1, t2, av, bv, sig, wtab1, wtab2);
    stage1_wmma_kernel<<<128, 512, 0, stream>>>(sig, wtab1, tvt, r1i);
    finalize_kernel<<<BATCH, 512, 0, stream>>>(r1i, wtab2, tvt, tvl, p, out);
}